// DMTetGeometry_78228534329468
// MI455X (gfx1250) — compile-verified
//
#include <hip/hip_runtime.h>
#include <hip/hip_bf16.h>
#include <math.h>

#define BLK 256

// TRI_TABLE packed 4 bits/entry, entry k in bits [4k+3:4k]; -1 encoded as 0xF.
__device__ __constant__ unsigned kTriPack[16] = {
    0x00FFFFFFu, 0x00FFF201u, 0x00FFF304u, 0x00431241u,
    0x00FFF513u, 0x00352032u, 0x00451041u, 0x00FFF524u,
    0x00FFF254u, 0x00154014u, 0x00253023u, 0x00FFF531u,
    0x00134214u, 0x00FFF403u, 0x00FFF102u, 0x00FFFFFFu
};

#if defined(__gfx1250__) && __has_builtin(__builtin_amdgcn_tensor_load_to_lds) && \
    __has_builtin(__builtin_amdgcn_s_wait_tensorcnt)
#define USE_TDM 1
#else
#define USE_TDM 0
#endif

typedef unsigned int v4u __attribute__((ext_vector_type(4)));
typedef int          v8i __attribute__((ext_vector_type(8)));
typedef int          v4i __attribute__((ext_vector_type(4)));

// Fast hardware reciprocal (v_rcp_f32, ~1 ulp) instead of the IEEE div-fixup
// chain; plenty of precision for an SDF lerp weight.
__device__ __forceinline__ float fast_rcp(float x) {
#if defined(__AMDGCN__) && __has_builtin(__builtin_amdgcn_rcpf)
  return __builtin_amdgcn_rcpf(x);
#else
  return 1.0f / x;
#endif
}

#if USE_TDM
// Issue a 1D TDM copy of `tile_elems` 4-byte elements from global `gsrc` into
// LDS at `lds_dst`. Reads past `tensor_elems` return zero (handles the partial
// last block). Tracked by TENSORcnt.
__device__ __forceinline__ void tdm_load_1d(void* lds_dst, const void* gsrc,
                                            unsigned tile_elems, unsigned tensor_elems) {
  unsigned long long ga = (unsigned long long)(uintptr_t)gsrc;
  // Generic pointers to LDS carry the LDS byte offset in their low 32 bits.
  unsigned lds_off = (unsigned)((unsigned long long)(uintptr_t)lds_dst & 0xFFFFFFFFull);

  v4u g0;
  g0.x = 1u;                                              // count=1 (valid user D#)
  g0.y = lds_off;                                         // lds_addr
  g0.z = (unsigned)(ga & 0xFFFFFFFFull);                  // global_addr[31:0]
  g0.w = ((unsigned)(ga >> 32) & 0x01FFFFFFu)             // global_addr[56:32]
         | 0x80000000u;                                   // type=2 ("image") in [127:126]

  v8i g1;
  g1[0] = 0x00020000;                                     // data_size=2 (4B); no mask/flags
  g1[1] = (int)((tensor_elems & 0xFFFFu) << 16);          // tensor_dim0[15:0]
  g1[2] = (int)((tensor_elems >> 16) & 0xFFFFu)           // tensor_dim0[31:16]
          | (1 << 16);                                    // tensor_dim1 = 1
  g1[3] = (int)(tile_elems << 16);                        // tile_dim0
  g1[4] = 0;                                              // tile_dim1=0, tile_dim2=0 (unused)
  g1[5] = (int)tile_elems;                                // tensor_dim0_stride (unused for 1 row)
  g1[6] = 0;
  g1[7] = 0;

  v4i z4 = {0, 0, 0, 0};
  v8i z8 = {0, 0, 0, 0, 0, 0, 0, 0};
  // 6-arg form (clang-23 / therock-10.0 headers): groups 2+3 + trailing v8i + cpol.
  __builtin_amdgcn_tensor_load_to_lds(g0, g1, z4, z4, z8, 0);
}
#endif

// ---------------------------------------------------------------------------
// Kernel 1: per-unique-edge crossing-vertex interpolation. verts = (Eu,3) f32.
// sdf (2.4MB) and pos (7.2MB) are L2-resident gather targets; verts is a
// write-once stream -> nontemporal stores.
// ---------------------------------------------------------------------------
__global__ void __launch_bounds__(BLK) edge_kernel(const float* __restrict__ pos,
                                                   const float* __restrict__ sdf,
                                                   const int* __restrict__ ue,
                                                   float* __restrict__ verts, int Eu) {
  int e = blockIdx.x * BLK + (int)threadIdx.x;
  if (e >= Eu) return;
  int2 ab = reinterpret_cast<const int2*>(ue)[e];         // coalesced 8B load
  float s0 = sdf[ab.x];
  float s1 = sdf[ab.y];
  bool cross = (s0 > 0.0f) != (s1 > 0.0f);
  float denom = cross ? (s0 - s1) : 1.0f;
  float rd = fast_rcp(denom);
  float w0 = -s1 * rd;
  float w1 = s0 * rd;
  float m = cross ? 1.0f : 0.0f;
  const float* p0 = pos + 3u * (size_t)(unsigned)ab.x;
  const float* p1 = pos + 3u * (size_t)(unsigned)ab.y;
  float* o = verts + 3u * (size_t)(unsigned)e;
  __builtin_nontemporal_store(m * fmaf(p0[0], w0, p1[0] * w1), o + 0);
  __builtin_nontemporal_store(m * fmaf(p0[1], w0, p1[1] * w1), o + 1);
  __builtin_nontemporal_store(m * fmaf(p0[2], w0, p1[2] * w1), o + 2);
}

// ---------------------------------------------------------------------------
// Kernel 2: per-tet marching-tets table -> faces (T,2,3) and uv_idx (T,2,3).
// The streaming tiles tet_fx4 (4KB/block) and idx_map (6KB/block) are staged
// into LDS by the Tensor Data Mover (wave 0 issues the D#s, waits TENSORcnt),
// keeping the VMEM queue free for the random sdf gathers.
// ---------------------------------------------------------------------------
__global__ void __launch_bounds__(BLK) tet_kernel(const float* __restrict__ sdf,
                                                  const int* __restrict__ tet,
                                                  const int* __restrict__ idx_map,
                                                  int* __restrict__ faces,
                                                  int* __restrict__ uv_idx, int T) {
  __shared__ int s_tet[BLK * 4];
  __shared__ int s_map[BLK * 6];
  const int bstart = (int)blockIdx.x * BLK;

#if USE_TDM
  if (threadIdx.x < 32) {
    unsigned remT4 = (unsigned)(T - bstart) * 4u;
    unsigned remT6 = (unsigned)(T - bstart) * 6u;
    tdm_load_1d(s_tet, tet + (size_t)bstart * 4, BLK * 4, remT4);
    tdm_load_1d(s_map, idx_map + (size_t)bstart * 6, BLK * 6, remT6);
    __builtin_amdgcn_s_wait_tensorcnt(0);
  }
  __syncthreads();
#else
  {
    int avail4 = T * 4 - bstart * 4; if (avail4 > BLK * 4) avail4 = BLK * 4;
    for (int k = (int)threadIdx.x; k < BLK * 4; k += BLK)
      s_tet[k] = (k < avail4) ? tet[(size_t)bstart * 4 + k] : 0;
    int avail6 = T * 6 - bstart * 6; if (avail6 > BLK * 6) avail6 = BLK * 6;
    for (int k = (int)threadIdx.x; k < BLK * 6; k += BLK)
      s_map[k] = (k < avail6) ? idx_map[(size_t)bstart * 6 + k] : 0;
    __syncthreads();
  }
#endif

  int i = bstart + (int)threadIdx.x;
  if (i >= T) return;
  int t = (int)threadIdx.x;

  int o0 = sdf[s_tet[4 * t + 0]] > 0.0f;
  int o1 = sdf[s_tet[4 * t + 1]] > 0.0f;
  int o2 = sdf[s_tet[4 * t + 2]] > 0.0f;
  int o3 = sdf[s_tet[4 * t + 3]] > 0.0f;
  int tidx = o0 | (o1 << 1) | (o2 << 2) | (o3 << 3);
  int sum = o0 + o1 + o2 + o3;
  // NUM_TRI_TABLE[tidx] == (popcount==2) ? 2 : (popcount==1||3) ? 1 : 0
  int ntri = (sum == 2) ? 2 : ((sum == 0 || sum == 4) ? 0 : 1);

  unsigned pack = kTriPack[tidx];
  size_t fb = (size_t)i * 6;
  int base = i * 4;                 // tet_idx == identity, so uv base = 4*i
#pragma unroll
  for (int k = 0; k < 6; ++k) {
    int trik = (int)((pack >> (4 * k)) & 0xFu); // 0..5 or 0xF (pad)
    int clipped = trik > 5 ? 5 : trik;          // jnp.clip(tri, 0, 5)
    int fk = s_map[6 * t + clipped];            // LDS-resident idx_map row
    int tri_j = k / 3;
    bool tv = tri_j < ntri;                     // tri_valid (implies surface tet)
    int r = k - tri_j * 3;                      // position within the triangle
    int uvv = (r == 0) ? base : base + tri_j + r;
    __builtin_nontemporal_store(tv ? fk : -1, faces + fb + k);
    __builtin_nontemporal_store(tv ? uvv : -1, uv_idx + fb + k);
  }
}

// ---------------------------------------------------------------------------
// Kernel 3: static UV atlas. One thread per grid cell writes 8 floats.
// linspace(0, 1-1/N, N)[k] == k/N exactly.
// ---------------------------------------------------------------------------
__global__ void __launch_bounds__(BLK) uv_kernel(float* __restrict__ uvs, int Ngrid) {
  int c = blockIdx.x * BLK + (int)threadIdx.x;
  if (c >= Ngrid * Ngrid) return;
  int i = c / Ngrid;
  int j = c - i * Ngrid;
  float inv = 1.0f / (float)Ngrid;
  float x = (float)j * inv;
  float y = (float)i * inv;
  float pad = 0.9f * inv;
  float* o = uvs + (size_t)c * 8;
  __builtin_nontemporal_store(x, o + 0);
  __builtin_nontemporal_store(y, o + 1);
  __builtin_nontemporal_store(x + pad, o + 2);
  __builtin_nontemporal_store(y, o + 3);
  __builtin_nontemporal_store(x + pad, o + 4);
  __builtin_nontemporal_store(y + pad, o + 5);
  __builtin_nontemporal_store(x, o + 6);
  __builtin_nontemporal_store(y + pad, o + 7);
}

extern "C" void kernel_launch(void* const* d_in, const int* in_sizes, int n_in,
                              void* d_out, int out_size, void* d_ws, size_t ws_size,
                              hipStream_t stream) {
  (void)n_in; (void)out_size; (void)d_ws; (void)ws_size;
  const float* pos     = (const float*)d_in[0];
  const float* sdf     = (const float*)d_in[1];
  const int*   tet     = (const int*)d_in[2];
  const int*   ue      = (const int*)d_in[3];
  const int*   idx_map = (const int*)d_in[4];

  const int Eu = in_sizes[3] / 2;
  const int T  = in_sizes[2] / 4;
  // Ngrid = ceil(sqrt((2T+1)//2)) = ceil(sqrt(T))
  int Ngrid = (int)floor(sqrt((double)T));
  while (Ngrid * Ngrid < T) ++Ngrid;
  const int G2 = Ngrid * Ngrid;

  // Outputs concatenated flat in return order (all 4-byte elements):
  // verts (Eu*3 f32) | faces (T*6 i32) | uvs (G2*8 f32) | uv_idx (T*6 i32)
  float* out_f  = (float*)d_out;
  float* verts  = out_f;
  int*   faces  = (int*)(out_f + (size_t)Eu * 3);
  float* uvs    = out_f + (size_t)Eu * 3 + (size_t)T * 6;
  int*   uv_idx = (int*)(out_f + (size_t)Eu * 3 + (size_t)T * 6 + (size_t)G2 * 8);

  edge_kernel<<<(Eu + BLK - 1) / BLK, BLK, 0, stream>>>(pos, sdf, ue, verts, Eu);
  tet_kernel<<<(T + BLK - 1) / BLK, BLK, 0, stream>>>(sdf, tet, idx_map, faces, uv_idx, T);
  uv_kernel<<<(G2 + BLK - 1) / BLK, BLK, 0, stream>>>(uvs, Ngrid);
}